// SparseRoiExtraCut_8358006358562
// MI455X (gfx1250) — compile-verified
//
#include <hip/hip_runtime.h>
#include <hip/hip_bf16.h>
#include <stdint.h>

// Problem constants (match reference)
#define NUM_BOXES     512
#define NUM_COORDS    200000
#define FEAT_C        128
#define MAX_SELECTED  262144
#define MASK_BYTES    (NUM_BOXES * NUM_COORDS)   // 102,400,000 bool bytes
#define THREADS       256
#define BPT           400                         // bytes per thread (25 x uint4)
#define CHUNK         (THREADS * BPT)             // 102,400 bytes per block
#define NB            (MASK_BYTES / CHUNK)        // 1000 blocks, exact

// ---- CDNA5 async global<->LDS path (gfx1250) ------------------------------
#if defined(__gfx1250__) && __has_builtin(__builtin_amdgcn_global_load_async_to_lds_b128) && __has_builtin(__builtin_amdgcn_global_store_async_from_lds_b128)
#define USE_ASYNC_LDS 1
#endif

// Pointer types the async builtins expect: v4i in AS1 (global) / AS3 (LDS)
typedef int v4i_ __attribute__((vector_size(16)));
typedef __attribute__((address_space(1))) v4i_ gv4i;
typedef __attribute__((address_space(3))) v4i_ lv4i;

__device__ __forceinline__ void wait_async0() {
#if defined(__gfx1250__) && __has_builtin(__builtin_amdgcn_s_wait_asynccnt)
  __builtin_amdgcn_s_wait_asynccnt(0);
#else
  asm volatile("s_wait_asynccnt 0" ::: "memory");
#endif
}

// Count nonzero bytes in a dword (robust even if bool bytes aren't 0/1)
__device__ __forceinline__ int nz_bytes(unsigned x) {
  unsigned m = (((x & 0x7F7F7F7Fu) + 0x7F7F7F7Fu) | x) & 0x80808080u;
  return __popc(m);
}

// ---- Kernel 1: per-thread + per-block counts ------------------------------
__global__ void srx_count(const unsigned char* __restrict__ mask,
                          int* __restrict__ threadCounts,
                          int* __restrict__ blockCounts) {
  const int b = blockIdx.x, t = threadIdx.x;
  const uint4* p = (const uint4*)(mask + (size_t)b * CHUNK + (size_t)t * BPT);
  int cnt = 0;
#pragma unroll
  for (int i = 0; i < BPT / 16; ++i) {
    uint4 v = p[i];
    cnt += nz_bytes(v.x) + nz_bytes(v.y) + nz_bytes(v.z) + nz_bytes(v.w);
  }
  threadCounts[b * THREADS + t] = cnt;
  __shared__ int s[THREADS];
  s[t] = cnt;
  __syncthreads();
  for (int off = THREADS / 2; off > 0; off >>= 1) {
    if (t < off) s[t] += s[t + off];
    __syncthreads();
  }
  if (t == 0) blockCounts[b] = s[0];
}

// ---- Kernel 2: exclusive scan of block counts (single block) --------------
__global__ void srx_scan(const int* __restrict__ blockCounts,
                         int* __restrict__ blockOffsets) {
  __shared__ int s[1024];
  const int t = threadIdx.x;
  int v = (t < NB) ? blockCounts[t] : 0;
  s[t] = v;
  __syncthreads();
  for (int off = 1; off < 1024; off <<= 1) {
    int add = (t >= off) ? s[t - off] : 0;
    __syncthreads();
    s[t] += add;
    __syncthreads();
  }
  if (t < NB) blockOffsets[t] = s[t] - v;   // exclusive
  if (t == NB - 1) blockOffsets[NB] = s[t]; // grand total
}

// ---- Kernel 3: ordered scatter of coord indices + extended coords ---------
__global__ void srx_scatter(const unsigned char* __restrict__ mask,
                            const int* __restrict__ coords,
                            const int* __restrict__ threadCounts,
                            const int* __restrict__ blockOffsets,
                            int* __restrict__ extOut,
                            int* __restrict__ coordIdx) {
  const int b = blockIdx.x, t = threadIdx.x;
  __shared__ int s[THREADS];
  const int tc = threadCounts[b * THREADS + t];
  s[t] = tc;
  __syncthreads();
  for (int off = 1; off < THREADS; off <<= 1) {
    int add = (t >= off) ? s[t - off] : 0;
    __syncthreads();
    s[t] += add;
    __syncthreads();
  }
  int pos = blockOffsets[b] + s[t] - tc; // global rank of this thread's first hit
  if (tc == 0) return;                   // no barriers below, safe to exit

  const int base = b * CHUNK + t * BPT;
  const uint4* p = (const uint4*)(mask + base);
  for (int i = 0; i < BPT / 16; ++i) {
    uint4 v = p[i];
    if ((v.x | v.y | v.z | v.w) == 0u) continue; // ~97% of groups are empty
    unsigned dw[4] = {v.x, v.y, v.z, v.w};
#pragma unroll
    for (int q = 0; q < 4; ++q) {
      unsigned x = dw[q];
      if (!x) continue;
#pragma unroll
      for (int jb = 0; jb < 4; ++jb) {
        if ((x >> (jb * 8)) & 0xFFu) {
          if (pos < MAX_SELECTED) {
            const int f = base + i * 16 + q * 4 + jb; // flat row-major index
            const int box = f / NUM_COORDS;
            const int c = f - box * NUM_COORDS;
            coordIdx[pos] = c;
            const int* cp = coords + (size_t)c * 4;
            int4 e;
            e.x = cp[0]; e.y = cp[1]; e.z = cp[2]; e.w = box;
            ((int4*)extOut)[pos] = e;
          }
          ++pos;
        }
      }
    }
  }
}

// ---- Kernel 4: feature-row gather via async LDS path + tail zeroing -------
// One wave32 per output row: 32 lanes x B128 = 512 B = one feature row.
__global__ void srx_gather(const float* __restrict__ features,
                           const int* __restrict__ coordIdx,
                           const int* __restrict__ blockOffsets,
                           float* __restrict__ outF,
                           int* __restrict__ extOut) {
  __shared__ __align__(16) float buf[THREADS / 32][FEAT_C];
  int total = blockOffsets[NB];
  if (total > MAX_SELECTED) total = MAX_SELECTED;
  const int wid = threadIdx.x >> 5;
  const int lane = threadIdx.x & 31;
  const int r = blockIdx.x * (THREADS / 32) + wid;
  if (r >= MAX_SELECTED) return;

  float* dst = outF + (size_t)r * FEAT_C + lane * 4;
  if (r < total) {
    const int c = coordIdx[r];
    const float* src = features + (size_t)c * FEAT_C + lane * 4;
#ifdef USE_ASYNC_LDS
    float* l = &buf[wid][lane * 4];
    __builtin_amdgcn_global_load_async_to_lds_b128((gv4i*)src, (lv4i*)l, 0, 0);
    wait_async0(); // LDS RAW: load complete before store reads it
    __builtin_amdgcn_global_store_async_from_lds_b128((gv4i*)dst, (lv4i*)l, 0, 0);
    // s_endpgm performs implicit wait-idle; store completion is covered.
#else
    float4 v = *(const float4*)src;
    *(float4*)dst = v;
#endif
  } else {
    // fill_value tail: zero both outputs (harness poisons d_out with 0xAA)
    float4 z = make_float4(0.f, 0.f, 0.f, 0.f);
    *(float4*)dst = z;
    if (lane == 0) {
      int4 zi = make_int4(0, 0, 0, 0);
      ((int4*)extOut)[r] = zi;
    }
  }
}

// ---- Launcher -------------------------------------------------------------
extern "C" void kernel_launch(void* const* d_in, const int* in_sizes, int n_in,
                              void* d_out, int out_size, void* d_ws, size_t ws_size,
                              hipStream_t stream) {
  const int* coords = (const int*)d_in[0];                   // (200000,4) int32
  const float* features = (const float*)d_in[1];             // (200000,128) f32
  const unsigned char* mask = (const unsigned char*)d_in[2]; // (512,200000) bool bytes
  (void)in_sizes; (void)n_in; (void)out_size; (void)ws_size;

  // Workspace layout (ints): threadCounts[NB*256] | blockCounts[NB] |
  // blockOffsets[NB+1] | coordIdx[MAX_SELECTED]  ->  ~2.1 MB total
  int* ws = (int*)d_ws;
  int* threadCounts = ws;                       // 256,000
  int* blockCounts  = ws + NB * THREADS;        // +256,000
  int* blockOffsets = blockCounts + NB;         // +257,000
  int* coordIdx     = blockOffsets + NB + 4;    // +258,004 (16B aligned)

  int* extOut = (int*)d_out;                              // 262144 x 4 int32
  float* outF = (float*)d_out + (size_t)MAX_SELECTED * 4; // 262144 x 128 f32

  srx_count<<<NB, THREADS, 0, stream>>>(mask, threadCounts, blockCounts);
  srx_scan<<<1, 1024, 0, stream>>>(blockCounts, blockOffsets);
  srx_scatter<<<NB, THREADS, 0, stream>>>(mask, coords, threadCounts,
                                          blockOffsets, extOut, coordIdx);
  srx_gather<<<MAX_SELECTED / (THREADS / 32), THREADS, 0, stream>>>(
      features, coordIdx, blockOffsets, outF, extOut);
}